// BSplineLayer_65910568125168
// MI455X (gfx1250) — compile-verified
//
#include <hip/hip_runtime.h>
#include <math.h>
#include <stdint.h>

// B-spline layer for MI455X (gfx1250).
//
// Roofline: elementwise cubic eval over 33.55M f32 + global min/max.
// Naive HBM traffic = 134MB (minmax read) + 134MB (eval read) + 134MB (write)
// = 402MB -> ~17us at 23.3 TB/s. But x (134MB) FITS in the 192MB L2:
//   pass1 reads x with default RT policy (populates L2),
//   pass3 re-reads x (L2 hits) and writes the output NON-TEMPORAL so the
//   store stream does not evict x. Effective HBM traffic ~268MB -> ~11.5us.
// Compute: ~15 VALU flops/elem (~44 TFLOP/s at BW saturation) < fp32 peak,
// so the kernel stays memory-bound. WMMA has no role (no matmul structure);
// the CDNA5 path used here is ASYNCcnt-tracked global->LDS B128 streaming,
// double-buffered per wave32 and pipelined with s_wait_asynccnt.

#define BLOCK 256           // 8 wave32s per block
#define WPB   8             // waves per block
#define NB1   512           // partial-reduction blocks
#define EVAL_BLOCKS 2048

typedef float __attribute__((ext_vector_type(4))) fvec4;

// ---------------- wave32 reductions ----------------
__device__ __forceinline__ float wave_min32(float v) {
#pragma unroll
  for (int off = 16; off > 0; off >>= 1)
    v = fminf(v, __shfl_xor(v, off, 32));
  return v;
}
__device__ __forceinline__ float wave_max32(float v) {
#pragma unroll
  for (int off = 16; off > 0; off >>= 1)
    v = fmaxf(v, __shfl_xor(v, off, 32));
  return v;
}

// ---------------- pass 1: per-block min/max partials ----------------
// Default (RT) loads on purpose: this pass pulls all of x into the 192MB L2
// so the eval pass can re-read it at L2 bandwidth.
__global__ void minmax_partial(const float* __restrict__ x, int n,
                               float* __restrict__ part) {
  float mn = INFINITY, mx = -INFINITY;
  const int stride = gridDim.x * blockDim.x;
  const int tid0   = blockIdx.x * blockDim.x + threadIdx.x;
  const int n4 = n >> 2;
  const fvec4* __restrict__ x4 = (const fvec4*)x;
  for (int i = tid0; i < n4; i += stride) {
    fvec4 v = x4[i];
    mn = fminf(mn, fminf(fminf(v.x, v.y), fminf(v.z, v.w)));
    mx = fmaxf(mx, fmaxf(fmaxf(v.x, v.y), fmaxf(v.z, v.w)));
  }
  for (int i = (n4 << 2) + tid0; i < n; i += stride) {  // tail (none for 2^25)
    float v = x[i];
    mn = fminf(mn, v);
    mx = fmaxf(mx, v);
  }
  mn = wave_min32(mn);
  mx = wave_max32(mx);
  __shared__ float smn[WPB], smx[WPB];
  const int wave = threadIdx.x >> 5;
  if ((threadIdx.x & 31) == 0) { smn[wave] = mn; smx[wave] = mx; }
  __syncthreads();
  if (threadIdx.x == 0) {
    float bmn = smn[0], bmx = smx[0];
#pragma unroll
    for (int w = 1; w < WPB; ++w) {
      bmn = fminf(bmn, smn[w]);
      bmx = fmaxf(bmx, smx[w]);
    }
    part[2 * blockIdx.x]     = bmn;
    part[2 * blockIdx.x + 1] = bmx;
  }
}

// ---------------- pass 2: final min/max (1 block) ----------------
__global__ void minmax_final(const float* __restrict__ part, int nblk,
                             float* __restrict__ mm) {
  float mn = INFINITY, mx = -INFINITY;
  for (int i = threadIdx.x; i < nblk; i += blockDim.x) {
    mn = fminf(mn, part[2 * i]);
    mx = fmaxf(mx, part[2 * i + 1]);
  }
  mn = wave_min32(mn);
  mx = wave_max32(mx);
  __shared__ float smn[WPB], smx[WPB];
  const int wave = threadIdx.x >> 5;
  if ((threadIdx.x & 31) == 0) { smn[wave] = mn; smx[wave] = mx; }
  __syncthreads();
  if (threadIdx.x == 0) {
    float bmn = smn[0], bmx = smx[0];
#pragma unroll
    for (int w = 1; w < WPB; ++w) {
      bmn = fminf(bmn, smn[w]);
      bmx = fmaxf(bmx, smx[w]);
    }
    mm[0] = bmn;
    mm[1] = bmx;
  }
}

// ---------------- pass 3: cubic de Boor eval (interval index fixed at 3) ----
// 256 floats (1KB) per wave chunk: 2x async B128 per lane per stage,
// double-buffered, s_wait_asynccnt-pipelined. Output stored non-temporal
// so the store stream does not evict the L2-resident x.
__global__ void bspline_eval_kernel(const float* __restrict__ x,
                                    const float* __restrict__ knots,
                                    const float* __restrict__ coeffs,
                                    const float* __restrict__ mm,
                                    float* __restrict__ out, int n) {
  const float mn   = mm[0];
  const float mx   = mm[1];
  const float invR = 1.0f / (mx - mn + 1e-8f);
  const float t1 = knots[1], t2 = knots[2], t3 = knots[3];
  const float t4 = knots[4], t5 = knots[5], t6 = knots[6];
  const float c0 = coeffs[0], c1 = coeffs[1], c2 = coeffs[2], c3 = coeffs[3];

  // alpha_{r,j} = ((x-mn)*invR - tl) / (tr - tl)  ==  fma(x, A, B)
  const float nb = -mn * invR;
#define MKAB(TL, TR, A, B)                \
  {                                       \
    float inv = 1.0f / ((TR) - (TL));     \
    A = invR * inv;                       \
    B = (nb - (TL)) * inv;                \
  }
  float A13, B13, A12, B12, A11, B11, A23, B23, A22, B22, A33, B33;
  MKAB(t3, t6, A13, B13)   // r=1, j=3
  MKAB(t2, t5, A12, B12)   // r=1, j=2
  MKAB(t1, t4, A11, B11)   // r=1, j=1
  MKAB(t3, t5, A23, B23)   // r=2, j=3
  MKAB(t2, t4, A22, B22)   // r=2, j=2
  MKAB(t3, t4, A33, B33)   // r=3, j=3
#undef MKAB
  const float e3 = c3 - c2, e2 = c2 - c1, e1 = c1 - c0;

  auto evalOne = [&](float xv) -> float {
    float a13 = fmaf(xv, A13, B13);
    float a12 = fmaf(xv, A12, B12);
    float a11 = fmaf(xv, A11, B11);
    float d3 = fmaf(a13, e3, c2);
    float d2 = fmaf(a12, e2, c1);
    float d1 = fmaf(a11, e1, c0);
    float a23 = fmaf(xv, A23, B23);
    float a22 = fmaf(xv, A22, B22);
    d3 = fmaf(a23, d3 - d2, d2);
    d2 = fmaf(a22, d2 - d1, d1);
    float a33 = fmaf(xv, A33, B33);
    return fmaf(a33, d3 - d2, d2);
  };
  auto eval4 = [&](fvec4 v) -> fvec4 {
    fvec4 r;
    r.x = evalOne(v.x);
    r.y = evalOne(v.y);
    r.z = evalOne(v.z);
    r.w = evalOne(v.w);
    return r;
  };

  // ---- async double-buffered streaming: 256 floats (1KB) per wave chunk ----
  __shared__ fvec4 tile[WPB][2][64];  // 16KB
  const int lane = threadIdx.x & 31;
  const int wave = threadIdx.x >> 5;
  const unsigned ldsA0 = (unsigned)(uintptr_t)&tile[wave][0][lane];
  const unsigned ldsA1 = (unsigned)(uintptr_t)&tile[wave][0][lane + 32];
  const unsigned ldsB0 = (unsigned)(uintptr_t)&tile[wave][1][lane];
  const unsigned ldsB1 = (unsigned)(uintptr_t)&tile[wave][1][lane + 32];

  const int nChunks    = n >> 8;           // 256 floats per chunk
  const int waveGlobal = blockIdx.x * WPB + wave;
  const int waveStride = gridDim.x * WPB;
  fvec4* __restrict__ out4 = (fvec4*)out;

  int c = waveGlobal;
  unsigned buf = 0;
  if (c < nChunks) {
    unsigned g0 = (unsigned)c * 1024u + (unsigned)lane * 16u;
    asm volatile("global_load_async_to_lds_b128 %0, %1, %2"
                 :: "v"(ldsA0), "v"(g0), "s"(x) : "memory");
    asm volatile("global_load_async_to_lds_b128 %0, %1, %2"
                 :: "v"(ldsA1), "v"(g0 + 512u), "s"(x) : "memory");
  }
  for (; c < nChunks; c += waveStride) {
    const int cn = c + waveStride;
    if (cn < nChunks) {
      unsigned g0 = (unsigned)cn * 1024u + (unsigned)lane * 16u;
      asm volatile("global_load_async_to_lds_b128 %0, %1, %2"
                   :: "v"(buf ? ldsA0 : ldsB0), "v"(g0), "s"(x) : "memory");
      asm volatile("global_load_async_to_lds_b128 %0, %1, %2"
                   :: "v"(buf ? ldsA1 : ldsB1), "v"(g0 + 512u), "s"(x) : "memory");
      asm volatile("s_wait_asynccnt 0x2" ::: "memory");  // current stage landed
    } else {
      asm volatile("s_wait_asynccnt 0x0" ::: "memory");
    }
    fvec4 v0 = tile[wave][buf][lane];        // ds_read_b128
    fvec4 v1 = tile[wave][buf][lane + 32];   // ds_read_b128
    fvec4 r0 = eval4(v0);
    fvec4 r1 = eval4(v1);
    const unsigned ob = (unsigned)c * 64u + (unsigned)lane;
    __builtin_nontemporal_store(r0, &out4[ob]);        // th:NT, keep x in L2
    __builtin_nontemporal_store(r1, &out4[ob + 32u]);
    buf ^= 1;
  }

  // tail (n % 256) — empty for 8192*4096 but kept for safety
  for (int i = (nChunks << 8) + blockIdx.x * blockDim.x + threadIdx.x; i < n;
       i += gridDim.x * blockDim.x) {
    float r = evalOne(x[i]);
    __builtin_nontemporal_store(r, &out[i]);
  }
}

// ---------------- host launch ----------------
extern "C" void kernel_launch(void* const* d_in, const int* in_sizes, int n_in,
                              void* d_out, int out_size, void* d_ws, size_t ws_size,
                              hipStream_t stream) {
  const float* x      = (const float*)d_in[0];
  const float* knots  = (const float*)d_in[1];
  const float* coeffs = (const float*)d_in[2];
  float* out = (float*)d_out;
  const int n = in_sizes[0];

  float* part = (float*)d_ws;        // 2*NB1 floats of partials
  float* mm   = part + 2 * NB1;      // [min, max]

  minmax_partial<<<NB1, BLOCK, 0, stream>>>(x, n, part);
  minmax_final<<<1, BLOCK, 0, stream>>>(part, NB1, mm);
  bspline_eval_kernel<<<EVAL_BLOCKS, BLOCK, 0, stream>>>(x, knots, coeffs, mm,
                                                         out, n);
}